// GNNStack_3229815406766
// MI455X (gfx1250) — compile-verified
//
#include <hip/hip_runtime.h>
#include <hip/hip_bf16.h>

typedef float v2f __attribute__((ext_vector_type(2)));
typedef float v8f __attribute__((ext_vector_type(8)));

#define DIN 128
#define XW  132   // R + DIN
#define DOUT 40

// ---------------------------------------------------------------------------
// init: split x into feats (N x 128) and role coefficients (N x 4):
//   coef = [2*roles0, roles1, roles2, 0]   (4th role coefficient is zero)
// ---------------------------------------------------------------------------
__global__ __launch_bounds__(256) void init_feats_coef(
    const float* __restrict__ x, float* __restrict__ feats,
    float* __restrict__ coef, int N)
{
    int i = blockIdx.x * blockDim.x + threadIdx.x;
    if (i < N * DIN) {
        int r = i / DIN, c = i % DIN;
        feats[i] = x[r * XW + 4 + c];
    }
    if (i < N) {
        coef[i * 4 + 0] = 2.0f * x[i * XW + 0];
        coef[i * 4 + 1] = x[i * XW + 1];
        coef[i * 4 + 2] = x[i * XW + 2];
        coef[i * 4 + 3] = 0.0f;
    }
}

// ---------------------------------------------------------------------------
// degree count: cnt[dst] += 1 per edge (identical across layers, run once)
// ---------------------------------------------------------------------------
__global__ __launch_bounds__(256) void count_edges(
    const int* __restrict__ dst, float* __restrict__ cnt, int E)
{
    int e = blockIdx.x * blockDim.x + threadIdx.x;
    if (e < E) atomicAdd(&cnt[dst[e]], 1.0f);
}

// ---------------------------------------------------------------------------
// Node-level message GEMM:
//   m[n] = relu( sum_{r<3} coef_r[n] * (feats[n] @ Wm_r + bm_r) )
// One block = 16 nodes x 128 cols; 8 waves, one 16x16 WMMA tile each.
// ---------------------------------------------------------------------------
__global__ __launch_bounds__(256) void msg_gemm(
    const float* __restrict__ feats,  // N x 128
    const float* __restrict__ coef,   // N x 4
    const float* __restrict__ Wm,     // 4 x 128 x 128 (layer slice)
    const float* __restrict__ bm,     // 4 x 128
    float* __restrict__ m_out,        // N x 128
    int N)
{
    __shared__ float sA[16 * DIN];
    __shared__ float sCoef[16 * 4];

    int row0 = blockIdx.x * 16;
    int tid  = threadIdx.x;

    for (int i = tid; i < 16 * DIN; i += 256) {
        int r = i / DIN, c = i % DIN;
        int gr = row0 + r;
        sA[i] = (gr < N) ? feats[gr * DIN + c] : 0.0f;
    }
    if (tid < 64) {
        int gr = row0 + (tid >> 2);
        sCoef[tid] = (gr < N) ? coef[gr * 4 + (tid & 3)] : 0.0f;
    }
    __syncthreads();

    int wave  = tid >> 5;
    int lane  = tid & 31;
    int col0  = wave * 16;
    int n     = col0 + (lane & 15);
    int khalf = (lane >> 4) * 2;      // 0 or 2
    int mrow  = lane & 15;

    v8f acc0 = {}, acc1 = {}, acc2 = {};
    for (int k = 0; k < DIN; k += 4) {
        v2f a;
        a.x = sA[mrow * DIN + k + khalf];
        a.y = sA[mrow * DIN + k + khalf + 1];
        v2f b0, b1, b2;
        const float* W0 = Wm;
        const float* W1 = Wm + DIN * DIN;
        const float* W2 = Wm + 2 * DIN * DIN;
        b0.x = W0[(k + khalf) * DIN + n];     b0.y = W0[(k + khalf + 1) * DIN + n];
        b1.x = W1[(k + khalf) * DIN + n];     b1.y = W1[(k + khalf + 1) * DIN + n];
        b2.x = W2[(k + khalf) * DIN + n];     b2.y = W2[(k + khalf + 1) * DIN + n];
        acc0 = __builtin_amdgcn_wmma_f32_16x16x4_f32(false, a, false, b0, (short)0, acc0, false, false);
        acc1 = __builtin_amdgcn_wmma_f32_16x16x4_f32(false, a, false, b1, (short)0, acc1, false, false);
        acc2 = __builtin_amdgcn_wmma_f32_16x16x4_f32(false, a, false, b2, (short)0, acc2, false, false);
    }

    int halfsel = (lane >> 4) * 8;
    float bi0 = bm[0 * DIN + n];
    float bi1 = bm[1 * DIN + n];
    float bi2 = bm[2 * DIN + n];
#pragma unroll
    for (int i = 0; i < 8; ++i) {
        int rrow = i + halfsel;
        int grow = row0 + rrow;
        if (grow < N) {
            float c0 = sCoef[rrow * 4 + 0];
            float c1 = sCoef[rrow * 4 + 1];
            float c2 = sCoef[rrow * 4 + 2];
            float v = c0 * (acc0[i] + bi0) + c1 * (acc1[i] + bi1) + c2 * (acc2[i] + bi2);
            m_out[grow * DIN + n] = fmaxf(v, 0.0f);
        }
    }
}

// ---------------------------------------------------------------------------
// Edge scatter: ssum[dst] += m[src].  One wave32 per edge, 4 floats per lane.
// 61.4M GLOBAL_ATOMIC_ADD_F32; ssum (20 MB) resides in L2.
// ---------------------------------------------------------------------------
__global__ __launch_bounds__(256) void edge_scatter(
    const int* __restrict__ src, const int* __restrict__ dst,
    const float* __restrict__ m, float* __restrict__ ssum, int E)
{
    int wid  = (blockIdx.x * blockDim.x + threadIdx.x) >> 5;
    int lane = threadIdx.x & 31;
    if (wid >= E) return;
    int s = src[wid];
    int d = dst[wid];
    const float4 v = *(const float4*)(m + (size_t)s * DIN + lane * 4);
    float* base = ssum + (size_t)d * DIN + lane * 4;
    atomicAdd(base + 0, v.x);
    atomicAdd(base + 1, v.y);
    atomicAdd(base + 2, v.z);
    atomicAdd(base + 3, v.w);
}

// ---------------------------------------------------------------------------
// Update GEMM (K=256, cat = [feats | ssum/cnt]) + role combine + bias + relu
// + fused row L2 normalization.
// ---------------------------------------------------------------------------
__global__ __launch_bounds__(256) void aggr_gemm(
    const float* __restrict__ feats, const float* __restrict__ ssum,
    const float* __restrict__ cnt, const float* __restrict__ coef,
    const float* __restrict__ Wa,   // 4 x 256 x 128 (layer slice)
    const float* __restrict__ ba,   // 4 x 128
    float* __restrict__ feats_out, int N)
{
    __shared__ float sA[16 * 256];     // 16 KB: cat tile
    __shared__ float sCoef[16 * 4];
    __shared__ float sOut[16 * DIN];   // 8 KB
    __shared__ float sNorm[16];

    int row0 = blockIdx.x * 16;
    int tid  = threadIdx.x;

    for (int i = tid; i < 16 * 256; i += 256) {
        int r = i >> 8, c = i & 255;
        int gr = row0 + r;
        float v = 0.0f;
        if (gr < N) {
            if (c < DIN) {
                v = feats[gr * DIN + c];
            } else {
                float cv = cnt[gr];
                v = (cv > 0.0f) ? ssum[gr * DIN + (c - DIN)] / cv : 0.0f;
            }
        }
        sA[i] = v;
    }
    if (tid < 64) {
        int gr = row0 + (tid >> 2);
        sCoef[tid] = (gr < N) ? coef[gr * 4 + (tid & 3)] : 0.0f;
    }
    __syncthreads();

    int wave  = tid >> 5;
    int lane  = tid & 31;
    int col0  = wave * 16;
    int n     = col0 + (lane & 15);
    int khalf = (lane >> 4) * 2;
    int mrow  = lane & 15;

    v8f acc0 = {}, acc1 = {}, acc2 = {};
    for (int k = 0; k < 256; k += 4) {
        v2f a;
        a.x = sA[mrow * 256 + k + khalf];
        a.y = sA[mrow * 256 + k + khalf + 1];
        v2f b0, b1, b2;
        const float* W0 = Wa;
        const float* W1 = Wa + 256 * DIN;
        const float* W2 = Wa + 2 * 256 * DIN;
        b0.x = W0[(k + khalf) * DIN + n];     b0.y = W0[(k + khalf + 1) * DIN + n];
        b1.x = W1[(k + khalf) * DIN + n];     b1.y = W1[(k + khalf + 1) * DIN + n];
        b2.x = W2[(k + khalf) * DIN + n];     b2.y = W2[(k + khalf + 1) * DIN + n];
        acc0 = __builtin_amdgcn_wmma_f32_16x16x4_f32(false, a, false, b0, (short)0, acc0, false, false);
        acc1 = __builtin_amdgcn_wmma_f32_16x16x4_f32(false, a, false, b1, (short)0, acc1, false, false);
        acc2 = __builtin_amdgcn_wmma_f32_16x16x4_f32(false, a, false, b2, (short)0, acc2, false, false);
    }

    int halfsel = (lane >> 4) * 8;
    float bi0 = ba[0 * DIN + n];
    float bi1 = ba[1 * DIN + n];
    float bi2 = ba[2 * DIN + n];
#pragma unroll
    for (int i = 0; i < 8; ++i) {
        int rrow = i + halfsel;
        float c0 = sCoef[rrow * 4 + 0];
        float c1 = sCoef[rrow * 4 + 1];
        float c2 = sCoef[rrow * 4 + 2];
        float v = c0 * (acc0[i] + bi0) + c1 * (acc1[i] + bi1) + c2 * (acc2[i] + bi2);
        sOut[rrow * DIN + n] = fmaxf(v, 0.0f);
    }
    __syncthreads();

    if (tid < 16) {
        float s = 0.0f;
        for (int j = 0; j < DIN; ++j) {
            float v = sOut[tid * DIN + j];
            s += v * v;
        }
        sNorm[tid] = fmaxf(sqrtf(s), 1e-12f);
    }
    __syncthreads();

    for (int i = tid; i < 16 * DIN; i += 256) {
        int r = i / DIN, c = i % DIN;
        int gr = row0 + r;
        if (gr < N) feats_out[gr * DIN + c] = fmaxf(sOut[i] / sNorm[r], 0.0f);
    }
}

// ---------------------------------------------------------------------------
// z = feats @ W1 + b1   (plain 128x128 GEMM, no activation)
// ---------------------------------------------------------------------------
__global__ __launch_bounds__(256) void z_gemm(
    const float* __restrict__ feats, const float* __restrict__ W1,
    const float* __restrict__ b1, float* __restrict__ z, int N)
{
    __shared__ float sA[16 * DIN];
    int row0 = blockIdx.x * 16;
    int tid  = threadIdx.x;
    for (int i = tid; i < 16 * DIN; i += 256) {
        int r = i / DIN, c = i % DIN;
        int gr = row0 + r;
        sA[i] = (gr < N) ? feats[gr * DIN + c] : 0.0f;
    }
    __syncthreads();

    int wave  = tid >> 5;
    int lane  = tid & 31;
    int n     = wave * 16 + (lane & 15);
    int khalf = (lane >> 4) * 2;
    int mrow  = lane & 15;

    v8f acc = {};
    for (int k = 0; k < DIN; k += 4) {
        v2f a, b;
        a.x = sA[mrow * DIN + k + khalf];
        a.y = sA[mrow * DIN + k + khalf + 1];
        b.x = W1[(k + khalf) * DIN + n];
        b.y = W1[(k + khalf + 1) * DIN + n];
        acc = __builtin_amdgcn_wmma_f32_16x16x4_f32(false, a, false, b, (short)0, acc, false, false);
    }

    int halfsel = (lane >> 4) * 8;
    float bi = b1[n];
#pragma unroll
    for (int i = 0; i < 8; ++i) {
        int grow = row0 + i + halfsel;
        if (grow < N) z[grow * DIN + n] = acc[i] + bi;
    }
}

// ---------------------------------------------------------------------------
// Head: logits = z @ W2 + b2  (128 -> 40, padded to 48) + row log-softmax.
// 3 waves / block (cols 0..47).
// ---------------------------------------------------------------------------
__global__ __launch_bounds__(96) void head_gemm(
    const float* __restrict__ z, const float* __restrict__ W2,
    const float* __restrict__ b2, float* __restrict__ out, int N)
{
    __shared__ float sA[16 * DIN];
    __shared__ float sL[16 * 48];
    __shared__ float sLse[16];

    int row0 = blockIdx.x * 16;
    int tid  = threadIdx.x;
    for (int i = tid; i < 16 * DIN; i += 96) {
        int r = i / DIN, c = i % DIN;
        int gr = row0 + r;
        sA[i] = (gr < N) ? z[gr * DIN + c] : 0.0f;
    }
    __syncthreads();

    int wave  = tid >> 5;            // 0..2
    int lane  = tid & 31;
    int n     = wave * 16 + (lane & 15);   // 0..47
    int khalf = (lane >> 4) * 2;
    int mrow  = lane & 15;
    bool valid_col = (n < DOUT);

    v8f acc = {};
    for (int k = 0; k < DIN; k += 4) {
        v2f a, b;
        a.x = sA[mrow * DIN + k + khalf];
        a.y = sA[mrow * DIN + k + khalf + 1];
        b.x = valid_col ? W2[(k + khalf) * DOUT + n] : 0.0f;
        b.y = valid_col ? W2[(k + khalf + 1) * DOUT + n] : 0.0f;
        acc = __builtin_amdgcn_wmma_f32_16x16x4_f32(false, a, false, b, (short)0, acc, false, false);
    }

    int halfsel = (lane >> 4) * 8;
    float bi = valid_col ? b2[n] : 0.0f;
#pragma unroll
    for (int i = 0; i < 8; ++i) {
        int rrow = i + halfsel;
        sL[rrow * 48 + n] = acc[i] + bi;
    }
    __syncthreads();

    if (tid < 16) {
        float mx = -3.4e38f;
        for (int j = 0; j < DOUT; ++j) mx = fmaxf(mx, sL[tid * 48 + j]);
        float s = 0.0f;
        for (int j = 0; j < DOUT; ++j) s += __expf(sL[tid * 48 + j] - mx);
        sLse[tid] = mx + __logf(s);
    }
    __syncthreads();

    for (int i = tid; i < 16 * DOUT; i += 96) {
        int r = i / DOUT, c = i % DOUT;
        int gr = row0 + r;
        if (gr < N) out[(size_t)gr * DOUT + c] = sL[r * 48 + c] - sLse[r];
    }
}

// ---------------------------------------------------------------------------
extern "C" void kernel_launch(void* const* d_in, const int* in_sizes, int n_in,
                              void* d_out, int out_size, void* d_ws, size_t ws_size,
                              hipStream_t stream)
{
    const float* x    = (const float*)d_in[0];
    const int*   ei   = (const int*)d_in[1];   // (2, E) int32
    const float* Wm   = (const float*)d_in[2]; // (2, 4, 128, 128)
    const float* bm   = (const float*)d_in[3]; // (2, 4, 128)
    const float* Wa   = (const float*)d_in[4]; // (2, 4, 256, 128)
    const float* ba   = (const float*)d_in[5]; // (2, 4, 128)
    const float* W1   = (const float*)d_in[6];
    const float* b1   = (const float*)d_in[7];
    const float* W2   = (const float*)d_in[8];
    const float* b2   = (const float*)d_in[9];
    float* out = (float*)d_out;

    const int N = in_sizes[0] / XW;       // 40000
    const int E = in_sizes[1] / 2;        // 480000
    const int* src = ei;
    const int* dst = ei + E;

    // workspace carve-up (256B aligned)
    char* base = (char*)d_ws;
    size_t off = 0;
    auto carve = [&](size_t bytes) -> float* {
        float* p = (float*)(base + off);
        off = (off + bytes + 255) & ~(size_t)255;
        return p;
    };
    float* coef   = carve((size_t)N * 4 * sizeof(float));
    float* cnt    = carve((size_t)N * sizeof(float));
    float* feats0 = carve((size_t)N * DIN * sizeof(float));
    float* feats1 = carve((size_t)N * DIN * sizeof(float));
    float* msg    = carve((size_t)N * DIN * sizeof(float));
    float* ssum   = carve((size_t)N * DIN * sizeof(float));
    (void)ws_size;

    const int ntiles = (N + 15) / 16;

    hipMemsetAsync(cnt, 0, (size_t)N * sizeof(float), stream);
    init_feats_coef<<<(N * DIN + 255) / 256, 256, 0, stream>>>(x, feats0, coef, N);
    count_edges<<<(E + 255) / 256, 256, 0, stream>>>(dst, cnt, E);

    float* fcur = feats0;
    float* fnext = feats1;
    for (int l = 0; l < 2; ++l) {
        const float* Wm_l = Wm + (size_t)l * 4 * DIN * DIN;
        const float* bm_l = bm + (size_t)l * 4 * DIN;
        const float* Wa_l = Wa + (size_t)l * 4 * 256 * DIN;
        const float* ba_l = ba + (size_t)l * 4 * DIN;

        msg_gemm<<<ntiles, 256, 0, stream>>>(fcur, coef, Wm_l, bm_l, msg, N);
        hipMemsetAsync(ssum, 0, (size_t)N * DIN * sizeof(float), stream);
        edge_scatter<<<(E + 7) / 8, 256, 0, stream>>>(src, dst, msg, ssum, E);
        aggr_gemm<<<ntiles, 256, 0, stream>>>(fcur, ssum, cnt, coef, Wa_l, ba_l, fnext, N);
        float* t = fcur; fcur = fnext; fnext = t;
    }

    z_gemm<<<ntiles, 256, 0, stream>>>(fcur, W1, b1, msg /*reuse as z*/, N);
    head_gemm<<<ntiles, 96, 0, stream>>>(msg, W2, b2, out, N);
}